// Model_45071386804718
// MI455X (gfx1250) — compile-verified
//
#include <hip/hip_runtime.h>
#include <math.h>

// Problem constants
constexpr int kB = 64;      // batch
constexpr int kL = 512;     // sequence length
constexpr int kD = 768;     // hidden dim
constexpr int kS = 8;       // segments

constexpr int APAD = 772;   // A-tile LDS row pitch in floats (768+4: conflict-free banks)

typedef __attribute__((ext_vector_type(2))) float v2f;
typedef __attribute__((ext_vector_type(8))) float v8f;

__device__ __forceinline__ float sigmoid_(float x) { return 1.0f / (1.0f + expf(-x)); }

// -------------------------------------------------------------------------
// Kernel 1: segment mean pooling for segments 5,6,7 only (all that matters).
// One block per (batch, segment); 192 threads x float4 = 768 floats per row.
// -------------------------------------------------------------------------
__global__ void seg_pool_kernel(const float* __restrict__ bert,
                                const int* __restrict__ segb,
                                float* __restrict__ S5,
                                float* __restrict__ S6,
                                float* __restrict__ S7) {
    int blk = blockIdx.x;
    int b = blk / 3;
    int s = 5 + (blk % 3);
    int start = segb[b * (kS + 1) + s];
    int end   = segb[b * (kS + 1) + s + 1];
    int tid = threadIdx.x;                       // 0..191

    float4 acc = make_float4(0.f, 0.f, 0.f, 0.f);
    for (int l = start; l < end; ++l) {
        const float* p = bert + ((size_t)b * kL + l) * kD + tid * 4;
        __builtin_prefetch(p + kD, 0, 1);        // global_prefetch_b8 of next row
        float4 v = *(const float4*)p;
        acc.x += v.x; acc.y += v.y; acc.z += v.z; acc.w += v.w;
    }
    float inv = 1.0f / (float)(end - start);
    acc.x *= inv; acc.y *= inv; acc.z *= inv; acc.w *= inv;
    float* outp = (s == 5 ? S5 : (s == 6 ? S6 : S7)) + (size_t)b * kD + tid * 4;
    *(float4*)outp = acc;
}

// -------------------------------------------------------------------------
// Kernel 2: multi-job GEMM. Each job: Y = tanh( sum_t X_t * W_t + bias ),
// M=64, N=768, K=768, exact fp32 via V_WMMA_F32_16X16X4_F32.
// B element (k,n) at W[k*rs + n*cs + off]:
//   dense weight (D,D): rs=D, cs=1,   off=0
//   conv weight (O,I,3): rs=3, cs=3D, off=tap   (B[k=i, n=o] = w[o,i,tap])
// Block = 256 thr (8 wave32). Block owns one M-tile (16 rows of X), staged
// into LDS with GLOBAL_LOAD_ASYNC_TO_LDS_B128 (ASYNCcnt) and shared by all
// 8 waves; each wave owns one N-tile. 24 blocks per job. All control flow is
// block-uniform and term handling fully unrolled (no runtime struct indexing
// -> no scratch spill, pointers stay in the global address space).
// -------------------------------------------------------------------------
struct Term {
    const float* X;
    const float* W;
    int rs, cs, off;
};
struct Job {
    float* Y;
    const float* bias;
    int nTerms;
    Term t[3];
};

__device__ __forceinline__ void stage_A(const float* __restrict__ X,
                                        unsigned ldsBase, int m0, int tid) {
    // 16 rows x 768 floats = 3072 float4, 128b per lane, fully coalesced
    #pragma unroll
    for (int it = 0; it < 12; ++it) {
        int idx = it * 256 + tid;
        int row = idx / 192;                     // 192 float4 per row
        int c4  = idx - row * 192;
        const float* g = X + (size_t)(m0 + row) * kD + c4 * 4;
        unsigned loff = ldsBase + row * (APAD * 4) + c4 * 16;
        asm volatile("global_load_async_to_lds_b128 %0, %1, off"
                     :: "v"(loff), "v"(g) : "memory");
    }
    asm volatile("s_wait_asynccnt 0x0" ::: "memory");
    __syncthreads();
}

__device__ __forceinline__ void kloop(v8f& acc, const float* __restrict__ W,
                                      int rs, int cs, int off,
                                      const float* __restrict__ Asf,
                                      int n0, int half, int l15) {
    const float* wcol = W + (size_t)(n0 + l15) * cs + off;  // B: lane = N col
    const float* arow = Asf + l15 * APAD;                   // A: lane = M row
    #pragma unroll 4
    for (int k = 0; k < kD; k += 4) {
        int kk = k + half * 2;       // lanes 0-15: K={k,k+1}; lanes 16-31: {k+2,k+3}
        v2f a = *(const v2f*)(arow + kk);                   // ds_load_b64 (padded, no conflicts)
        v2f bf;
        bf.x = wcol[(size_t)kk * rs];
        bf.y = wcol[(size_t)(kk + 1) * rs];
        acc = __builtin_amdgcn_wmma_f32_16x16x4_f32(
            /*neg_a=*/false, a, /*neg_b=*/false, bf,
            /*c_mod=*/(short)0, acc, /*reuse_a=*/false, /*reuse_b=*/false);
    }
}

__device__ __forceinline__ void gemm_job(const Job& J, const float* Asf,
                                         unsigned ldsBase, int m0, int n0,
                                         int tid, int half, int l15) {
    v8f acc = {};
    stage_A(J.t[0].X, ldsBase, m0, tid);
    kloop(acc, J.t[0].W, J.t[0].rs, J.t[0].cs, J.t[0].off, Asf, n0, half, l15);
    if (J.nTerms > 1) {                          // nTerms is block-uniform
        __syncthreads();                         // all waves done reading old A
        stage_A(J.t[1].X, ldsBase, m0, tid);
        kloop(acc, J.t[1].W, J.t[1].rs, J.t[1].cs, J.t[1].off, Asf, n0, half, l15);
    }
    if (J.nTerms > 2) {
        __syncthreads();
        stage_A(J.t[2].X, ldsBase, m0, tid);
        kloop(acc, J.t[2].W, J.t[2].rs, J.t[2].cs, J.t[2].off, Asf, n0, half, l15);
    }
    int col = n0 + l15;
    float bv = J.bias[col];
    #pragma unroll
    for (int r = 0; r < 8; ++r) {                // D layout: VGPR r -> M = r (+8 hi lanes)
        int row = m0 + r + half * 8;
        J.Y[(size_t)row * kD + col] = tanhf(acc[r] + bv);
    }
}

__global__ __launch_bounds__(256) void wmma_gemm_tanh(Job j0, Job j1, Job j2) {
    __shared__ float Asf[16 * APAD];             // 49,408 B of the 320 KB LDS
    int tid  = threadIdx.x;
    int wave = tid >> 5;
    int lane = tid & 31;
    int half = lane >> 4;
    int l15  = lane & 15;

    int jb    = blockIdx.x / 24;                 // job index (block-uniform)
    int local = blockIdx.x % 24;
    int m0 = (local & 3) * 16;                   // 4 M-tiles, shared A stage
    int n0 = ((local >> 2) * 8 + wave) * 16;     // 6 n-groups x 8 waves = 48 N-tiles

    // low 32 bits of a flat LDS pointer == wave-relative LDS byte offset
    unsigned ldsBase = (unsigned)(size_t)(&Asf[0]);

    if (jb == 0)      gemm_job(j0, Asf, ldsBase, m0, n0, tid, half, l15);
    else if (jb == 1) gemm_job(j1, Asf, ldsBase, m0, n0, tid, half, l15);
    else              gemm_job(j2, Asf, ldsBase, m0, n0, tid, half, l15);
}

// -------------------------------------------------------------------------
// Kernel 3: per-batch hop refinement (last segment only), softmax gate, output.
// One block (256 threads, 8 wave32s) per batch; each thread owns 3 elements.
// -------------------------------------------------------------------------
__device__ __forceinline__ float block_reduce_sum(float v, float* sred) {
    #pragma unroll
    for (int o = 16; o > 0; o >>= 1) v += __shfl_down(v, o, 32);
    int lane = threadIdx.x & 31, w = threadIdx.x >> 5;
    if (lane == 0) sred[w] = v;
    __syncthreads();
    float r = (threadIdx.x < 8) ? sred[threadIdx.x] : 0.0f;
    if (w == 0) {
        #pragma unroll
        for (int o = 4; o > 0; o >>= 1) r += __shfl_down(r, o, 32);
        if (lane == 0) sred[8] = r;
    }
    __syncthreads();
    float out = sred[8];
    __syncthreads();
    return out;
}

__global__ void finalize_kernel(const float* __restrict__ S7,
                                const float* __restrict__ H2,
                                const float* __restrict__ C27,
                                const float* __restrict__ ln_g,
                                const float* __restrict__ ln_b,
                                const float* __restrict__ target,
                                const float* __restrict__ fc2_w,
                                const float* __restrict__ fc2_b,
                                float* __restrict__ out) {
    __shared__ float sred[16];
    int b = blockIdx.x;
    int tid = threadIdx.x;

    float s7v[3], tg[3], lg[3], lb[3];
    #pragma unroll
    for (int j = 0; j < 3; ++j) {
        int i = tid + j * 256;
        s7v[j] = S7[(size_t)b * kD + i];
        tg[j]  = target[i];          // target[0, :] (first batch row)
        lg[j]  = ln_g[i];
        lb[j]  = ln_b[i];
    }
    float ss = block_reduce_sum(s7v[0]*s7v[0] + s7v[1]*s7v[1] + s7v[2]*s7v[2], sred);

    float abr[3][3];                 // [branch][local element]
    for (int br = 0; br < 3; ++br) {
        float cur[3];
        #pragma unroll
        for (int j = 0; j < 3; ++j) {
            int i = tid + j * 256;
            if (br == 0)      cur[j] = H2[(size_t)b * kD + i];   // a3 init: g2[:, -1, :]
            else if (br == 1) cur[j] = C27[(size_t)b * kD + i];  // a4 init: conv_out[:, -1, :]
            else              cur[j] = sigmoid_(ss * s7v[j]);    // a5 init: tmask[:, -1, :]
        }
        for (int hop = 0; hop < 2; ++hop) {                      // HOP - 1 iterations
            float c = block_reduce_sum(cur[0]*s7v[0] + cur[1]*s7v[1] + cur[2]*s7v[2], sred);
            float t[3], sum_p = 0.f, sq_p = 0.f;
            #pragma unroll
            for (int j = 0; j < 3; ++j) {
                t[j] = sigmoid_(c * s7v[j]);
                sum_p += t[j];
                sq_p  += t[j] * t[j];
            }
            float sum = block_reduce_sum(sum_p, sred);
            float sq  = block_reduce_sum(sq_p,  sred);
            float m   = sum * (1.0f / kD);
            float var = sq * (1.0f / kD) - m * m;
            float rinv = rsqrtf(var + 1e-5f);
            #pragma unroll
            for (int j = 0; j < 3; ++j)                          // LAMB * LN + sentence
                cur[j] = 0.5f * ((t[j] - m) * rinv * lg[j] + lb[j]) + s7v[j];
        }
        abr[br][0] = cur[0]; abr[br][1] = cur[1]; abr[br][2] = cur[2];
    }

    // softmax gate over the 3 branches against target[0]
    float d0 = block_reduce_sum(abr[0][0]*tg[0] + abr[0][1]*tg[1] + abr[0][2]*tg[2], sred);
    float d1 = block_reduce_sum(abr[1][0]*tg[0] + abr[1][1]*tg[1] + abr[1][2]*tg[2], sred);
    float d2 = block_reduce_sum(abr[2][0]*tg[0] + abr[2][1]*tg[1] + abr[2][2]*tg[2], sred);
    float mx = fmaxf(d0, fmaxf(d1, d2));
    float e0 = expf(d0 - mx), e1 = expf(d1 - mx), e2 = expf(d2 - mx);
    float inv = 1.0f / (e0 + e1 + e2);
    float wk[3] = {e0 * inv, e1 * inv, e2 * inv};

    // out[b, j] = sum_k wk[k] * (a_k . fc2_w[k*D:(k+1)*D, j]) + fc2_b[j]
    for (int j = 0; j < 3; ++j) {
        float part = 0.f;
        #pragma unroll
        for (int k = 0; k < 3; ++k) {
            float p = 0.f;
            #pragma unroll
            for (int q = 0; q < 3; ++q) {
                int i = tid + q * 256;
                p += abr[k][q] * fc2_w[(size_t)(k * kD + i) * 3 + j];
            }
            part += wk[k] * p;
        }
        float tot = block_reduce_sum(part, sred);
        if (tid == 0) out[b * 3 + j] = tot + fc2_b[j];
    }
}

// -------------------------------------------------------------------------
// Host launch: 4 launches total (pool -> stage1 x3 jobs -> stage2 x2 jobs -> finalize)
// -------------------------------------------------------------------------
extern "C" void kernel_launch(void* const* d_in, const int* in_sizes, int n_in,
                              void* d_out, int out_size, void* d_ws, size_t ws_size,
                              hipStream_t stream) {
    const float* bert    = (const float*)d_in[0];
    const float* target  = (const float*)d_in[1];
    const float* gc1_w   = (const float*)d_in[2];
    const float* gc1_b   = (const float*)d_in[3];
    const float* gc2_w   = (const float*)d_in[4];
    const float* gc2_b   = (const float*)d_in[5];
    const float* conv1_w = (const float*)d_in[6];
    const float* conv1_b = (const float*)d_in[7];
    const float* conv2_w = (const float*)d_in[8];
    const float* conv2_b = (const float*)d_in[9];
    const float* ln_g    = (const float*)d_in[10];
    const float* ln_b    = (const float*)d_in[11];
    const float* fc2_w   = (const float*)d_in[12];
    const float* fc2_b   = (const float*)d_in[13];
    const int*   segb    = (const int*)d_in[14];
    (void)in_sizes; (void)n_in; (void)out_size; (void)ws_size;

    float* ws = (float*)d_ws;
    const size_t MAT = (size_t)kB * kD;
    float* S5  = ws;
    float* S6  = S5  + MAT;
    float* S7  = S6  + MAT;
    float* H1  = S7  + MAT;
    float* C16 = H1  + MAT;
    float* C17 = C16 + MAT;
    float* H2  = C17 + MAT;
    float* C27 = H2  + MAT;

    Term z{nullptr, nullptr, 0, 0, 0};

    seg_pool_kernel<<<kB * 3, 192, 0, stream>>>(bert, segb, S5, S6, S7);

    // Stage 1 (independent jobs):
    //   H1  = tanh(S7 @ gc1_w + gc1_b)                       (g1[:, -1, :])
    //   C16 = tanh(S5*W1k0 + S6*W1k1 + S7*W1k2 + b1)         (conv1 @ pos 6)
    //   C17 = tanh(S6*W1k0 + S7*W1k1 + b1)                   (conv1 @ pos 7, k2 pad)
    Job s1a{H1,  gc1_b,   1, {Term{S7, gc1_w, kD, 1, 0}, z, z}};
    Job s1b{C16, conv1_b, 3, {Term{S5, conv1_w, 3, 3 * kD, 0},
                              Term{S6, conv1_w, 3, 3 * kD, 1},
                              Term{S7, conv1_w, 3, 3 * kD, 2}}};
    Job s1c{C17, conv1_b, 2, {Term{S6, conv1_w, 3, 3 * kD, 0},
                              Term{S7, conv1_w, 3, 3 * kD, 1}, z}};
    wmma_gemm_tanh<<<72, 256, 0, stream>>>(s1a, s1b, s1c);

    // Stage 2 (independent jobs):
    //   H2  = tanh(H1 @ gc2_w + gc2_b)                       (g2[:, -1, :])
    //   C27 = tanh(C16*W2k0 + C17*W2k1 + b2)                 (conv2 @ pos 7, k2 pad)
    Job s2a{H2,  gc2_b,   1, {Term{H1, gc2_w, kD, 1, 0}, z, z}};
    Job s2b{C27, conv2_b, 2, {Term{C16, conv2_w, 3, 3 * kD, 0},
                              Term{C17, conv2_w, 3, 3 * kD, 1}, z}};
    wmma_gemm_tanh<<<48, 256, 0, stream>>>(s2a, s2b, s2a);

    finalize_kernel<<<kB, 256, 0, stream>>>(S7, H2, C27, ln_g, ln_b,
                                            target, fc2_w, fc2_b, (float*)d_out);
}